// TaylorFeatureMap_54941221651114
// MI455X (gfx1250) — compile-verified
//
#include <hip/hip_runtime.h>

typedef __attribute__((ext_vector_type(2))) float v2f;
typedef __attribute__((ext_vector_type(8))) float v8f;

#define HEAD_DIM 64
#define ROW_OUT  2145   // 1 + 64 + 64 + 2016

// One wave32 per input row. Off-diagonal outer-product terms via
// v_wmma_f32_16x16x4_f32 (exact f32 rank-1 product; B pre-scaled by 1/8,
// which is an exact power-of-2 scale, so results match fl(x_i*x_j)/8).
__global__ __launch_bounds__(256) void TaylorFeatureMap_kernel(
    const float* __restrict__ x, float* __restrict__ out, int nrows) {
  const int lane = threadIdx.x & 31;
  const int wave = threadIdx.x >> 5;
  const int row  = blockIdx.x * 8 + wave;
  if (row >= nrows) return;  // uniform per wave: EXEC all-1s at every WMMA

  const float* __restrict__ xr   = x   + (size_t)row * HEAD_DIM;
  float* __restrict__       orow = out + (size_t)row * ROW_OUT;

  const float c_lin  = 0.35355339059327384f;  // 1 / 64^0.25
  const float c_diag = 0.08838834764831845f;  // 1 / (8*sqrt(2))

  // Coalesced row load: lane l gets x[l] and x[l+32]
  const float xlo = xr[lane];
  const float xhi = xr[lane + 32];

  // Head of output row: [1, x/rrd (64), x*x/(rd*sqrt2) (64)]
  if (lane == 0) __builtin_nontemporal_store(1.0f, orow);
  __builtin_nontemporal_store(xlo * c_lin,        orow + 1  + lane);
  __builtin_nontemporal_store(xhi * c_lin,        orow + 33 + lane);
  __builtin_nontemporal_store(xlo * xlo * c_diag, orow + 65 + lane);
  __builtin_nontemporal_store(xhi * xhi * c_diag, orow + 97 + lane);

  // Broadcast 16-element blocks so lanes 0..15 hold block b's values.
  const int l15 = lane & 15;
  const int mhi = (lane >> 4) << 3;  // 0 for lanes 0-15, 8 for lanes 16-31
  const float s0 = __shfl(xlo, l15,      32);
  const float s1 = __shfl(xlo, l15 + 16, 32);
  const float s2 = __shfl(xhi, l15,      32);
  const float s3 = __shfl(xhi, l15 + 16, 32);

  // A (16x4 f32): K=0 column lives in lanes 0-15 of VGPR0; rest zero.
  // B (4x16 f32): K=0 row lives in lanes 0-15 of VGPR0; rest zero.
  // B is pre-scaled by 1/8 (exact), folding the offdiag normalization
  // into the WMMA and removing all per-element scale multiplies.
  const bool lo16 = (lane < 16);
  v2f a[4], b[4];
  a[0] = (v2f){lo16 ? s0 : 0.0f, 0.0f};
  a[1] = (v2f){lo16 ? s1 : 0.0f, 0.0f};
  a[2] = (v2f){lo16 ? s2 : 0.0f, 0.0f};
  a[3] = (v2f){lo16 ? s3 : 0.0f, 0.0f};
  b[0] = (v2f){a[0].x * 0.125f, 0.0f};
  b[1] = (v2f){a[1].x * 0.125f, 0.0f};
  b[2] = (v2f){a[2].x * 0.125f, 0.0f};
  b[3] = (v2f){a[3].x * 0.125f, 0.0f};

  // Per-lane column pointer; row bases g(i) become compile-time literal
  // pairs (lane-half select), and the 16*bj column step becomes an
  // immediate store offset.
  float* __restrict__ offlane = orow + 129 + l15;
  const int hm = l15 - mhi;  // diag tile: store iff hm > r

#pragma unroll
  for (int bi = 0; bi < 4; ++bi) {
    v8f acc[4];
#pragma unroll
    for (int bj = bi; bj < 4; ++bj) {
      v8f c = {};
      // D[m][n] = x[16*bi+m] * x[16*bj+n] / 8   (exact f32)
      acc[bj] = __builtin_amdgcn_wmma_f32_16x16x4_f32(
          /*neg_a=*/false, a[bi], /*neg_b=*/false, b[bj],
          /*c_mod=*/(short)0, c, /*reuse_a=*/false, /*reuse_b=*/false);
    }
#pragma unroll
    for (int r = 0; r < 8; ++r) {
      const int i0 = 16 * bi + r;        // row index for lanes 0-15
      const int i1 = i0 + 8;             // row index for lanes 16-31
      const int g0 = (125 * i0 - i0 * i0) / 2 - 1;  // compile-time literal
      const int g1 = (125 * i1 - i1 * i1) / 2 - 1;  // compile-time literal
      float* p = offlane + (lo16 ? g0 : g1);
      // diagonal tile: strict upper triangle only
      if (hm > r) __builtin_nontemporal_store(acc[bi][r], p + 16 * bi);
#pragma unroll
      for (int bj = bi + 1; bj < 4; ++bj)
        __builtin_nontemporal_store(acc[bj][r], p + 16 * bj);
    }
  }
}

extern "C" void kernel_launch(void* const* d_in, const int* in_sizes, int n_in,
                              void* d_out, int out_size, void* d_ws, size_t ws_size,
                              hipStream_t stream) {
  const float* x = (const float*)d_in[0];
  float* out = (float*)d_out;
  const int nrows = in_sizes[0] / HEAD_DIM;   // 131072 for the reference shapes
  const int waves_per_block = 8;              // 256 threads, wave32
  const int blocks = (nrows + waves_per_block - 1) / waves_per_block;
  hipLaunchKernelGGL(TaylorFeatureMap_kernel, dim3(blocks), dim3(256), 0, stream,
                     x, out, nrows);
}